// GCN_4501125726314
// MI455X (gfx1250) — compile-verified
//
#include <hip/hip_runtime.h>
#include <cstdint>
#include <cstddef>

// ---------------------------------------------------------------------------
// GCN forward on MI455X (gfx1250).
// GEMMs via V_WMMA_F32_16X16X4_F32, B operands staged in LDS (ds_load_2addr_b64),
// weights pre-transposed + zero-padded so the inner loop is branch-free.
// Workload is scatter/bandwidth bound -> keep full f32 precision.
// ---------------------------------------------------------------------------

constexpr int   kN    = 50000;
constexpr int   kE    = 600000;
constexpr int   kD    = 128;     // D_IN == D_HID
constexpr int   kDOUT = 40;
constexpr float kEPS  = 1e-5f;

typedef __attribute__((ext_vector_type(2))) float v2f;
typedef __attribute__((ext_vector_type(8))) float v8f;

// ---------------------------- small utility kernels ------------------------

__global__ void k_fill1(float* __restrict__ p, int n) {
  int i = blockIdx.x * blockDim.x + threadIdx.x;
  if (i < n) p[i] = 1.0f;
}

__global__ void k_zero4(float4* __restrict__ p, long long n4) {
  long long i = (long long)blockIdx.x * blockDim.x + threadIdx.x;
  if (i < n4) p[i] = make_float4(0.f, 0.f, 0.f, 0.f);
}

__global__ void k_deg_accum(const int* __restrict__ dst, float* __restrict__ deg) {
  int e = blockIdx.x * blockDim.x + threadIdx.x;
  if (e < kE) atomicAdd(&deg[dst[e]], 1.0f);
}

// deg (in place) -> dinv ; self_norm = dinv^2
__global__ void k_dinv(float* __restrict__ dinv_io, float* __restrict__ selfn) {
  int i = blockIdx.x * blockDim.x + threadIdx.x;
  if (i < kN) {
    float d = rsqrtf(dinv_io[i]);
    dinv_io[i] = d;
    selfn[i]   = d * d;
  }
}

__global__ void k_edge_norm(const int* __restrict__ src, const int* __restrict__ dst,
                            const float* __restrict__ dinv, float* __restrict__ norm) {
  int e = blockIdx.x * blockDim.x + threadIdx.x;
  if (e < kE) norm[e] = dinv[src[e]] * dinv[dst[e]];
}

// Transpose + zero-pad weights: Wt[c*kD + k] = (c < ncr) ? W[k*ncr + c] : 0.
__global__ void k_wtranspose(const float* __restrict__ W, float* __restrict__ Wt,
                             int ncols_real, int ncols_pad) {
  int i = blockIdx.x * blockDim.x + threadIdx.x;
  if (i < ncols_pad * kD) {
    int c = i / kD;
    int k = i - c * kD;
    Wt[i] = (c < ncols_real) ? W[(size_t)k * ncols_real + c] : 0.0f;
  }
}

// --------------------------- BatchNorm helpers -----------------------------

// Column sums / sums-of-squares over kN rows; blockDim.x == kD (=128).
__global__ void k_colstats(const float* __restrict__ x,
                           float* __restrict__ ssum, float* __restrict__ ssq) {
  const int c = threadIdx.x;
  float s = 0.0f, q = 0.0f;
  for (int r = blockIdx.x; r < kN; r += gridDim.x) {
    float v = x[(size_t)r * kD + c];
    s += v;
    q += v * v;
  }
  atomicAdd(&ssum[c], s);
  atomicAdd(&ssq[c], q);
}

// Fused GCN epilogue: z = agg + h*self_norm + bias (written back to agg),
// while accumulating BN column stats for the next normalization.
__global__ void k_epi_stats(float* __restrict__ z, const float* __restrict__ h,
                            const float* __restrict__ selfn, const float* __restrict__ bias,
                            float* __restrict__ ssum, float* __restrict__ ssq) {
  const int c = threadIdx.x;
  const float b = bias[c];
  float s = 0.0f, q = 0.0f;
  for (int r = blockIdx.x; r < kN; r += gridDim.x) {
    size_t i = (size_t)r * kD + c;
    float v = fmaf(h[i], selfn[r], z[i]) + b;
    z[i] = v;
    s += v;
    q += v * v;
  }
  atomicAdd(&ssum[c], s);
  atomicAdd(&ssq[c], q);
}

// mean/var -> per-column affine (scale, shift); one block, kD threads.
__global__ void k_bn_finalize(const float* __restrict__ ssum, const float* __restrict__ ssq,
                              const float* __restrict__ g, const float* __restrict__ b,
                              float* __restrict__ scale, float* __restrict__ shift) {
  const int c = threadIdx.x;
  const float invN = 1.0f / (float)kN;
  float mu  = ssum[c] * invN;
  float var = ssq[c] * invN - mu * mu;
  float sc  = g[c] * rsqrtf(var + kEPS);
  scale[c] = sc;
  shift[c] = b[c] - mu * sc;
}

template <bool RELU>
__global__ void k_bn_apply4(const float4* __restrict__ x, const float* __restrict__ scale,
                            const float* __restrict__ shift, float4* __restrict__ y) {
  long long i = (long long)blockIdx.x * blockDim.x + threadIdx.x;  // over ND/4
  if (i < (long long)kN * kD / 4) {
    int c = (int)((i * 4) & (kD - 1));
    float4 xv = x[i];
    float4 r;
    r.x = fmaf(xv.x, scale[c + 0], shift[c + 0]);
    r.y = fmaf(xv.y, scale[c + 1], shift[c + 1]);
    r.z = fmaf(xv.z, scale[c + 2], shift[c + 2]);
    r.w = fmaf(xv.w, scale[c + 3], shift[c + 3]);
    if (RELU) {
      r.x = fmaxf(r.x, 0.f); r.y = fmaxf(r.y, 0.f);
      r.z = fmaxf(r.z, 0.f); r.w = fmaxf(r.w, 0.f);
    }
    y[i] = r;
  }
}

// ------------------------------ edge scatter -------------------------------

// One wave per edge; each lane moves float4 (global_load_b128 + 4 f32 atomics).
__global__ void k_scatter(const float* __restrict__ h, const int* __restrict__ src,
                          const int* __restrict__ dst, const float* __restrict__ norm,
                          float* __restrict__ agg) {
  const int gid = blockIdx.x * blockDim.x + threadIdx.x;
  const int e   = gid >> 5;
  const int l   = gid & 31;
  if (e >= kE) return;
  const int s   = src[e];
  const int d   = dst[e];
  const float w = norm[e];
  const float4 hv = *(const float4*)&h[(size_t)s * kD + l * 4];
  float* ap = &agg[(size_t)d * kD + l * 4];
  atomicAdd(ap + 0, hv.x * w);
  atomicAdd(ap + 1, hv.y * w);
  atomicAdd(ap + 2, hv.z * w);
  atomicAdd(ap + 3, hv.w * w);
}

// ------------------------------ WMMA GEMM ----------------------------------
// C[N x DOUT] = A[N x 128] @ W[128 x DOUT] (+ bias), f32 via 16x16x4 WMMA.
// Wt is the transposed, zero-padded weight (Wt[col*kD + k], col < NT*16),
// cooperatively staged into LDS; inner loop is branch-free.
//
// V_WMMA_F32_16X16X4_F32 operand layouts (wave32):
//   A 16x4 : lane<16 -> row=lane,  {K=kk,  K=kk+1}; lane>=16 -> {K=kk+2, kk+3}
//   B 4x16 : lane<16 -> col=lane,  {K=kk,  K=kk+1}; lane>=16 -> {K=kk+2, kk+3}
//   C/D    : lanes 0-15 col=lane / lanes 16-31 col=lane-16; VGPR v -> M=v(+8)
template <int NT, int DOUT, bool ADD_BIAS>
__global__ void __launch_bounds__(256)
k_gemm_wmma(const float* __restrict__ A, const float* __restrict__ Wt,
            const float* __restrict__ bias, float* __restrict__ C) {
  constexpr int SW_ELEMS = NT * 16 * kD;         // <= 16384 floats (64 KB)
  __shared__ float sW[SW_ELEMS];

  // cooperative float4 staging of Wt (already transposed + padded)
  for (int i = threadIdx.x * 4; i < SW_ELEMS; i += 256 * 4) {
    *(float4*)&sW[i] = *(const float4*)&Wt[i];
  }
  __syncthreads();

  const int lane = threadIdx.x & 31;
  const int wid  = (blockIdx.x * blockDim.x + threadIdx.x) >> 5;  // wave-uniform
  if (wid >= kN / 16) return;                                     // uniform exit: EXEC stays full

  const int half = lane >> 4;   // 0 or 1
  const int l16  = lane & 15;
  const float* __restrict__ Arow = A + (size_t)(wid * 16 + l16) * kD;

  v8f acc[NT] = {};

#pragma unroll 2
  for (int kk = 0; kk < kD; kk += 4) {
    const int kb = kk + 2 * half;                 // even -> 8B aligned
    const v2f a = *(const v2f*)&Arow[kb];         // global_load_b64, coalesced
#pragma unroll
    for (int t = 0; t < NT; ++t) {
      const int col = t * 16 + l16;
      const v2f b = *(const v2f*)&sW[col * kD + kb];   // ds_load_2addr_b64 (paired)
      acc[t] = __builtin_amdgcn_wmma_f32_16x16x4_f32(
          /*neg_a=*/false, a, /*neg_b=*/false, b,
          /*c_mod=*/(short)0, acc[t], /*reuse_a=*/false, /*reuse_b=*/false);
    }
  }

#pragma unroll
  for (int t = 0; t < NT; ++t) {
    const int col = t * 16 + l16;
    if (col < DOUT) {                              // compile-time true for DOUT=128
      if constexpr (ADD_BIAS) {
        const float bv = bias[col];
#pragma unroll
        for (int v = 0; v < 8; ++v) {
          const int m = v + 8 * half;
          C[(size_t)(wid * 16 + m) * DOUT + col] = acc[t][v] + bv;
        }
      } else {
#pragma unroll
        for (int v = 0; v < 8; ++v) {
          const int m = v + 8 * half;
          C[(size_t)(wid * 16 + m) * DOUT + col] = acc[t][v];   // no +0.0 dead VALU
        }
      }
    }
  }
}

// ------------------------------ launcher -----------------------------------

static inline int cdiv_ll(long long a, long long b) { return (int)((a + b - 1) / b); }

extern "C" void kernel_launch(void* const* d_in, const int* in_sizes, int n_in,
                              void* d_out, int out_size, void* d_ws, size_t ws_size,
                              hipStream_t stream) {
  (void)in_sizes; (void)n_in; (void)out_size; (void)ws_size;

  const float* node_feat = (const float*)d_in[0];
  const int*   esrc      = (const int*)  d_in[1];
  const int*   edst      = (const int*)  d_in[2];
  const float* bn0_g     = (const float*)d_in[3];
  const float* bn0_b     = (const float*)d_in[4];
  const float* W1        = (const float*)d_in[5];
  const float* b1        = (const float*)d_in[6];
  const float* bn1_g     = (const float*)d_in[7];
  const float* bn1_b     = (const float*)d_in[8];
  const float* W2        = (const float*)d_in[9];
  const float* b2        = (const float*)d_in[10];
  const float* bn2_g     = (const float*)d_in[11];
  const float* bn2_b     = (const float*)d_in[12];
  const float* Wf        = (const float*)d_in[13];
  const float* bf        = (const float*)d_in[14];
  float*       out       = (float*)d_out;

  // workspace carve-up (floats): ~80 MB total; all offsets 16B-aligned.
  float* ws = (float*)d_ws;
  size_t o = 0;
  float* dinv  = ws + o; o += kN;
  float* selfn = ws + o; o += kN;
  float* enorm = ws + o; o += kE;
  float* ssum  = ws + o; o += kD;   // ssum & ssq contiguous (zeroed together)
  float* ssq   = ws + o; o += kD;
  float* scl   = ws + o; o += kD;
  float* shf   = ws + o; o += kD;
  float* wt1   = ws + o; o += (size_t)kD * kD;        // W1^T
  float* wt2   = ws + o; o += (size_t)kD * kD;        // W2^T
  float* wtf   = ws + o; o += (size_t)48 * kD;        // Wf^T zero-padded to 48 cols
  float* bufX  = ws + o; o += (size_t)kN * kD;        // GEMM input
  float* bufH  = ws + o; o += (size_t)kN * kD;        // GEMM output h
  float* bufA  = ws + o; o += (size_t)kN * kD;        // aggregation / z

  const int T = 256;
  const long long ND  = (long long)kN * kD;
  const long long ND4 = ND / 4;
  const int gemm_blocks    = cdiv_ll(kN / 16, 8);     // 8 waves (256 thr) per block
  const int scatter_blocks = cdiv_ll((long long)kE * 32, T);

  // ---- weight transpose/pad (tiny) -------------------------------------
  k_wtranspose<<<cdiv_ll(kD * kD, T), T, 0, stream>>>(W1, wt1, kD, kD);
  k_wtranspose<<<cdiv_ll(kD * kD, T), T, 0, stream>>>(W2, wt2, kD, kD);
  k_wtranspose<<<cdiv_ll(48 * kD, T), T, 0, stream>>>(Wf, wtf, kDOUT, 48);

  // ---- degrees / symmetric normalization -------------------------------
  k_fill1    <<<cdiv_ll(kN, T), T, 0, stream>>>(dinv, kN);
  k_deg_accum<<<cdiv_ll(kE, T), T, 0, stream>>>(edst, dinv);
  k_dinv     <<<cdiv_ll(kN, T), T, 0, stream>>>(dinv, selfn);
  k_edge_norm<<<cdiv_ll(kE, T), T, 0, stream>>>(esrc, edst, dinv, enorm);

  // ---- BN(input) --------------------------------------------------------
  k_zero4      <<<1, 64, 0, stream>>>((float4*)ssum, (2 * kD) / 4);
  k_colstats   <<<512, kD, 0, stream>>>(node_feat, ssum, ssq);
  k_bn_finalize<<<1, kD, 0, stream>>>(ssum, ssq, bn0_g, bn0_b, scl, shf);
  k_bn_apply4<false><<<cdiv_ll(ND4, T), T, 0, stream>>>((const float4*)node_feat, scl, shf,
                                                        (float4*)bufX);

  // ---- conv1: h = X@W1 ; agg ; z = agg + h*selfn + b1 ; BN ; ReLU ------
  k_gemm_wmma<8, kD, false><<<gemm_blocks, 256, 0, stream>>>(bufX, wt1, nullptr, bufH);
  k_zero4    <<<cdiv_ll(ND4, T), T, 0, stream>>>((float4*)bufA, ND4);
  k_scatter  <<<scatter_blocks, T, 0, stream>>>(bufH, esrc, edst, enorm, bufA);
  k_zero4    <<<1, 64, 0, stream>>>((float4*)ssum, (2 * kD) / 4);
  k_epi_stats<<<512, kD, 0, stream>>>(bufA, bufH, selfn, b1, ssum, ssq);
  k_bn_finalize<<<1, kD, 0, stream>>>(ssum, ssq, bn1_g, bn1_b, scl, shf);
  k_bn_apply4<true><<<cdiv_ll(ND4, T), T, 0, stream>>>((const float4*)bufA, scl, shf,
                                                       (float4*)bufX);

  // ---- conv2 ------------------------------------------------------------
  k_gemm_wmma<8, kD, false><<<gemm_blocks, 256, 0, stream>>>(bufX, wt2, nullptr, bufH);
  k_zero4    <<<cdiv_ll(ND4, T), T, 0, stream>>>((float4*)bufA, ND4);
  k_scatter  <<<scatter_blocks, T, 0, stream>>>(bufH, esrc, edst, enorm, bufA);
  k_zero4    <<<1, 64, 0, stream>>>((float4*)ssum, (2 * kD) / 4);
  k_epi_stats<<<512, kD, 0, stream>>>(bufA, bufH, selfn, b2, ssum, ssq);
  k_bn_finalize<<<1, kD, 0, stream>>>(ssum, ssq, bn2_g, bn2_b, scl, shf);
  k_bn_apply4<true><<<cdiv_ll(ND4, T), T, 0, stream>>>((const float4*)bufA, scl, shf,
                                                       (float4*)bufX);

  // ---- final projection: out = X@Wf + bf (DOUT=40, padded to 48) -------
  k_gemm_wmma<3, kDOUT, true><<<gemm_blocks, 256, 0, stream>>>(bufX, wtf, bf, out);
}